// Fuzzyfier_68719476858
// MI455X (gfx1250) — compile-verified
//
#include <hip/hip_runtime.h>
#include <cstdint>

// Fuzzyfier: out[b,v,s,p] = alpha_cut(exp(-(x[b,v,s]-c[v,p])^2 / (2*sigma[v,p]^2)))
// B=64 V=8 S=1024 P=64 -> 33.5M f32 outputs (128 MB). Store-bandwidth bound.
// Strategy: wave32 covers two s-rows per iteration (half-wave each), 4 p per
// lane -> b128 stores (512 B contiguous per wave-store). x tile staged to LDS
// via CDNA5 async load-to-LDS; inner-loop x reads are half-wave LDS broadcasts.

#define B_      64
#define V_      8
#define S_      1024
#define P_      64
#define SCHUNK  256            // s-rows handled per block
#define NCHUNK  (S_ / SCHUNK)  // 4
#define ALPHA   0.1f
#define LOG2E   1.4426950408889634f

__global__ __launch_bounds__(256) void Fuzzyfier_kernel(
    const float* __restrict__ x,        // (B, V, S)
    const float* __restrict__ fuzzy,    // (V, P, 2) interleaved [c, sigma]
    float* __restrict__ out)            // (B, V, S, P)
{
    __shared__ float xs[SCHUNK];

    const int tid   = threadIdx.x;
    const int lane  = tid & 31;
    const int wave  = tid >> 5;
    const int half  = lane >> 4;        // 0: lanes 0-15, 1: lanes 16-31
    const int l16   = lane & 15;

    const int chunk = blockIdx.x & (NCHUNK - 1);
    const int bv    = blockIdx.x >> 2;          // (b*V + v)
    const int v     = bv & (V_ - 1);
    const int s0    = chunk * SCHUNK;

    // ---- Stage x[bv, s0 : s0+SCHUNK] into LDS via CDNA5 async load-to-LDS.
    {
        const float* xsrc   = x + (size_t)bv * S_ + s0;
        uint32_t     ldsoff = (uint32_t)(uintptr_t)(&xs[tid]); // low 32 bits = LDS byte offset
        uint32_t     voff   = (uint32_t)(tid * 4);
        asm volatile("global_load_async_to_lds_b32 %0, %1, %2"
                     :: "v"(ldsoff), "v"(voff), "s"(xsrc)
                     : "memory");
        asm volatile("s_wait_asynccnt 0" ::: "memory");
    }
    __syncthreads();

    // ---- Per-lane Gaussian params for p = 4*l16 .. 4*l16+3 (two b128 loads).
    const float4 csA = *(const float4*)(fuzzy + (size_t)v * (P_ * 2) + l16 * 8);
    const float4 csB = *(const float4*)(fuzzy + (size_t)v * (P_ * 2) + l16 * 8 + 4);
    const float c0 = csA.x, c1 = csA.z, c2 = csB.x, c3 = csB.z;
    // Fold log2(e) so the loop uses bare v_exp_f32 (2^x): exp(-d^2/(2s^2)) = 2^(d^2 * g)
    const float g0 = -LOG2E / (2.0f * csA.y * csA.y);
    const float g1 = -LOG2E / (2.0f * csA.w * csA.w);
    const float g2 = -LOG2E / (2.0f * csB.y * csB.y);
    const float g3 = -LOG2E / (2.0f * csB.w * csB.w);

    // Wave w owns rows s = s0 + 32*w + (2*i + half); lane writes p = 4*l16..+3.
    float4* obase = (float4*)(out + ((size_t)bv * S_ + s0 + wave * 32) * P_)
                    + (size_t)half * (P_ / 4) + l16;
    const int xbase = wave * 32 + half;

    #pragma unroll 4
    for (int i = 0; i < 16; ++i) {
        const float xv = xs[xbase + 2 * i];   // half-wave-uniform LDS broadcast
        const float d0 = xv - c0;
        const float d1 = xv - c1;
        const float d2 = xv - c2;
        const float d3 = xv - c3;
        const float m0 = __builtin_amdgcn_exp2f(d0 * d0 * g0);
        const float m1 = __builtin_amdgcn_exp2f(d1 * d1 * g1);
        const float m2 = __builtin_amdgcn_exp2f(d2 * d2 * g2);
        const float m3 = __builtin_amdgcn_exp2f(d3 * d3 * g3);
        float4 r;
        r.x = (m0 >= ALPHA) ? m0 : 0.0f;
        r.y = (m1 >= ALPHA) ? m1 : 0.0f;
        r.z = (m2 >= ALPHA) ? m2 : 0.0f;
        r.w = (m3 >= ALPHA) ? m3 : 0.0f;
        obase[(size_t)i * 2 * (P_ / 4)] = r;  // global_store_b128, 512B per wave-store
    }
}

extern "C" void kernel_launch(void* const* d_in, const int* in_sizes, int n_in,
                              void* d_out, int out_size, void* d_ws, size_t ws_size,
                              hipStream_t stream) {
    (void)in_sizes; (void)n_in; (void)d_ws; (void)ws_size; (void)out_size;
    const float* x     = (const float*)d_in[0];   // (64, 8, 1024) f32
    const float* fuzzy = (const float*)d_in[1];   // (8, 64, 2)   f32
    float*       out   = (float*)d_out;           // (64, 8, 1024, 64) f32

    const int grid = B_ * V_ * NCHUNK;            // 2048 blocks
    Fuzzyfier_kernel<<<grid, 256, 0, stream>>>(x, fuzzy, out);
}